// ProteinBackboneFeature_12841952215193
// MI455X (gfx1250) — compile-verified
//
#include <hip/hip_runtime.h>
#include <math.h>

typedef float v2f  __attribute__((ext_vector_type(2)));
typedef float v8f  __attribute__((ext_vector_type(8)));
typedef unsigned int u32x4 __attribute__((ext_vector_type(4)));
typedef int   i32x8 __attribute__((ext_vector_type(8)));
typedef int   i32x4 __attribute__((ext_vector_type(4)));

#define NBEADS 128
#define NPAIR  8128            // 128*127/2
#define NANG   126
#define NDIH   125
#define FEAT   (NPAIR + NANG + 2 * NDIH)   // 8504 floats per batch

// One block per batch; 256 threads = 8 wave32s.
// Wave w computes Gram row-tile w against col-tiles w..7 via V_WMMA_F32_16X16X4_F32.
__global__ __launch_bounds__(256)
void protein_feat_kernel(const float* __restrict__ data, float* __restrict__ out)
{
    __shared__ float sRaw[NBEADS * 3];  // TDM landing zone (AoS xyz)
    __shared__ float sP[4][NBEADS];     // x, y, z, 0-pad rows (branch-free K=4 operands)
    __shared__ float sR[NBEADS];        // squared norms
    __shared__ float sF[FEAT];          // staged output (scatter in LDS, coalesce to HBM)

    const int b   = blockIdx.x;
    const int tid = threadIdx.x;
    const float* __restrict__ X = data + (size_t)b * (NBEADS * 3);
    float* __restrict__ O       = out  + (size_t)b * FEAT;

    // ---------------- TDM: async-tensor DMA of the 384-float block into LDS ----------------
    if (tid < 32) {   // wave 0 issues the tensor op (EXEC ignored; TENSORcnt is per-wave)
        unsigned long long gaddr = (unsigned long long)(uintptr_t)X;
        unsigned lds_addr = (unsigned)(uintptr_t)(&sRaw[0]);   // low 32 bits = LDS byte offset

        u32x4 g0;
        g0[0] = 1u;                                            // count=1, user descriptor
        g0[1] = lds_addr;                                      // D#.lds_addr
        g0[2] = (unsigned)(gaddr & 0xffffffffu);               // global_addr[31:0]  (bits 95:64)
        g0[3] = ((unsigned)(gaddr >> 32) & 0x01ffffffu)        // global_addr[56:32] (bits 120:96)
              | (2u << 30);                                    // type = 2 ("image")

        i32x8 g1;
        g1[0] = 0x00020000;          // workgroup_mask=0, data_size=2 (4 bytes)
        g1[1] = (int)(384u << 16);   // tensor_dim0 = 384 elements (bits 79:48, low half)
        g1[2] = (int)(1u << 16);     // tensor_dim1 = 1            (bits 111:80, low half)
        g1[3] = (int)(384u << 16);   // tile_dim0   = 384          (bits 127:112)
        g1[4] = 1;                   // tile_dim1   = 1            (bits 143:128)
        g1[5] = 384;                 // tensor_dim0_stride = 384   (bits 207:160, low 32)
        g1[6] = 0;
        g1[7] = 0;

        i32x4 g2 = {0, 0, 0, 0};
        i32x4 g3 = {0, 0, 0, 0};
        i32x8 g4 = {0, 0, 0, 0, 0, 0, 0, 0};   // 6-arg toolchain variant: extra group

        __builtin_amdgcn_tensor_load_to_lds(g0, g1, g2, g3, g4, 0);
        __builtin_amdgcn_s_wait_tensorcnt(0);
    }
    __syncthreads();

    // Transpose AoS -> SoA with zero-padded 4th row; squared norms.
    if (tid < NBEADS) {
        float x = sRaw[3 * tid + 0];
        float y = sRaw[3 * tid + 1];
        float z = sRaw[3 * tid + 2];
        sP[0][tid] = x;
        sP[1][tid] = y;
        sP[2][tid] = z;
        sP[3][tid] = 0.0f;
        sR[tid] = x * x + y * y + z * z;
    }
    __syncthreads();

    const int wave = tid >> 5;        // 0..7
    const int lane = tid & 31;
    const int lhi  = lane >> 4;       // 0 or 1 (lane half)
    const int llo  = lane & 15;

    // ---------------- Pairwise distances via WMMA Gram matrix ----------------
    // A-matrix 16x4 f32 layout: V0 = {K=0 | K=2}, V1 = {K=1 | K=3} across lane halves.
    // B-matrix 4x16 layout mirrors it; since B = X^T tile, operand bits are identical.
    const int ti = wave;
    const int ak = lhi << 1;          // 0 -> K{0,1}, 1 -> K{2,3}
    v2f A;
    A[0] = sP[ak + 0][ti * 16 + llo];
    A[1] = sP[ak + 1][ti * 16 + llo];

    // Row norms: i = ti*16 + v + 8*lhi is invariant across tj -> hoist.
    float ri[8];
#pragma unroll
    for (int v = 0; v < 8; ++v) ri[v] = sR[ti * 16 + v + (lhi << 3)];

    // --- Diagonal tile (tj == ti): partial, keep the j > i predicate ---
    {
        v8f C = {};
        C = __builtin_amdgcn_wmma_f32_16x16x4_f32(false, A, false, A,
                                                  (short)0, C, false, false);
        float rj = sR[ti * 16 + llo];
#pragma unroll
        for (int v = 0; v < 8; ++v) {
            int i = ti * 16 + v + (lhi << 3);
            int j = ti * 16 + llo;
            if (j > i) {
                float d2  = ri[v] + rj - 2.0f * C[v];
                float d   = __builtin_amdgcn_sqrtf(fmaxf(d2, 0.0f));
                int   off = j - i;
                int   idx = (off - 1) * NBEADS - (((off - 1) * off) >> 1) + i;
                sF[idx] = d;
            }
        }
    }

    // --- Off-diagonal tiles (tj > ti): every element valid, branch-free ---
    for (int tj = ti + 1; tj < 8; ++tj) {
        v2f Bv;
        Bv[0] = sP[ak + 0][tj * 16 + llo];
        Bv[1] = sP[ak + 1][tj * 16 + llo];
        v8f C = {};
        C = __builtin_amdgcn_wmma_f32_16x16x4_f32(false, A, false, Bv,
                                                  (short)0, C, false, false);
        float rj = sR[tj * 16 + llo];
        int   j  = tj * 16 + llo;
#pragma unroll
        for (int v = 0; v < 8; ++v) {
            int   i   = ti * 16 + v + (lhi << 3);
            float d2  = ri[v] + rj - 2.0f * C[v];
            float d   = __builtin_amdgcn_sqrtf(fmaxf(d2, 0.0f));
            int   off = j - i;
            int   idx = (off - 1) * NBEADS - (((off - 1) * off) >> 1) + i;
            sF[idx] = d;
        }
    }

    // ---------------- Angles (126) ----------------
    if (tid < NANG) {
        int t = tid;
        float a0x = sP[0][t + 1] - sP[0][t];
        float a0y = sP[1][t + 1] - sP[1][t];
        float a0z = sP[2][t + 1] - sP[2][t];
        float a1x = sP[0][t + 2] - sP[0][t + 1];
        float a1y = sP[1][t + 2] - sP[1][t + 1];
        float a1z = sP[2][t + 2] - sP[2][t + 1];
        float dp  = a0x * a1x + a0y * a1y + a0z * a1z;
        float n0  = __builtin_amdgcn_sqrtf(a0x * a0x + a0y * a0y + a0z * a0z);
        float n1  = __builtin_amdgcn_sqrtf(a1x * a1x + a1y * a1y + a1z * a1z);
        sF[NPAIR + t] = acosf(dp / (n0 * n1));
    }

    // ---------------- Dihedral cos/sin (125 each) ----------------
    if (tid < NDIH) {
        int t = tid;
        float p0x = sP[0][t],     p0y = sP[1][t],     p0z = sP[2][t];
        float p1x = sP[0][t + 1], p1y = sP[1][t + 1], p1z = sP[2][t + 1];
        float p2x = sP[0][t + 2], p2y = sP[1][t + 2], p2z = sP[2][t + 2];
        float p3x = sP[0][t + 3], p3y = sP[1][t + 3], p3z = sP[2][t + 3];

        float a0x = p1x - p0x, a0y = p1y - p0y, a0z = p1z - p0z;
        float a1x = p2x - p1x, a1y = p2y - p1y, a1z = p2z - p1z;
        float a2x = p3x - p2x, a2y = p3y - p2y, a2z = p3z - p2z;

        // c0 = a0 x a1, c1 = a1 x a2, plane = c1 x a1
        float c0x = a0y * a1z - a0z * a1y;
        float c0y = a0z * a1x - a0x * a1z;
        float c0z = a0x * a1y - a0y * a1x;
        float c1x = a1y * a2z - a1z * a2y;
        float c1y = a1z * a2x - a1x * a2z;
        float c1z = a1x * a2y - a1y * a2x;
        float plx = c1y * a1z - c1z * a1y;
        float ply = c1z * a1x - c1x * a1z;
        float plz = c1x * a1y - c1y * a1x;

        float nc0 = __builtin_amdgcn_sqrtf(c0x * c0x + c0y * c0y + c0z * c0z);
        float nc1 = __builtin_amdgcn_sqrtf(c1x * c1x + c1y * c1y + c1z * c1z);
        float npl = __builtin_amdgcn_sqrtf(plx * plx + ply * ply + plz * plz);

        float dcos = (c0x * c1x + c0y * c1y + c0z * c1z) / (nc0 * nc1);
        float dsin = (c0x * plx + c0y * ply + c0z * plz) / (nc0 * npl);

        sF[NPAIR + NANG + t]        = dcos;
        sF[NPAIR + NANG + NDIH + t] = dsin;
    }

    __syncthreads();

    // ---------------- Coalesced b128 write-out: the bandwidth-critical part ----------------
    // FEAT = 8504 floats = exactly 2126 float4s; 34016-byte batch stride is 16B-aligned.
    const float4* __restrict__ src = (const float4*)&sF[0];
    float4* __restrict__ dst       = (float4*)O;
#pragma unroll 2
    for (int k = tid; k < FEAT / 4; k += 256) {
        dst[k] = src[k];
    }
}

extern "C" void kernel_launch(void* const* d_in, const int* in_sizes, int n_in,
                              void* d_out, int out_size, void* d_ws, size_t ws_size,
                              hipStream_t stream)
{
    const float* data = (const float*)d_in[0];
    float*       out  = (float*)d_out;
    const int nbatch  = in_sizes[0] / (NBEADS * 3);   // 4096
    protein_feat_kernel<<<nbatch, 256, 0, stream>>>(data, out);
}